// SIRModel_2645699854679
// MI455X (gfx1250) — compile-verified
//
#include <hip/hip_runtime.h>

#define N_NODES 50000
#define N_EDGES 800000
#define DIM 128
#define NEG_SLOPE 0.2f

typedef float v2f __attribute__((ext_vector_type(2)));
typedef float v8f __attribute__((ext_vector_type(8)));
typedef int   v4i __attribute__((ext_vector_type(4)));

#if __has_builtin(__builtin_amdgcn_global_load_async_to_lds_b128) && \
    __has_builtin(__builtin_amdgcn_s_wait_asynccnt)
#define USE_ASYNC_LDS 1
#else
#define USE_ASYNC_LDS 0
#endif

__device__ __forceinline__ float leaky(float x) {
    return x >= 0.0f ? x : NEG_SLOPE * x;
}

// ---------------------------------------------------------------------------
// Utility fill
// ---------------------------------------------------------------------------
__global__ void fill_kernel(float* __restrict__ p, int n, float v) {
    int i = blockIdx.x * blockDim.x + threadIdx.x;
    if (i < n) p[i] = v;
}

// ---------------------------------------------------------------------------
// In-degree (as float) via L2-side atomics
// ---------------------------------------------------------------------------
__global__ void deg_kernel(const int* __restrict__ dst, float* __restrict__ deg) {
    int e = blockIdx.x * blockDim.x + threadIdx.x;
    if (e < N_EDGES) atomicAdd(&deg[dst[e]], 1.0f);
}

__global__ void invdenom_kernel(float* __restrict__ deg) {
    int n = blockIdx.x * blockDim.x + threadIdx.x;
    if (n < N_NODES) deg[n] = 1.0f / fmaxf(deg[n], 1.0f);
}

// ---------------------------------------------------------------------------
// Edge messages: agg[dst] += leaky(hq[dst] + hk[src]).
// One wave per edge (grid is exact, no divergence), edge indices forced
// scalar via readfirstlane -> s_load path; each lane handles a float4 of the
// 512B feature row (coalesced), then 4x global_atomic_add_f32 into L2.
// ---------------------------------------------------------------------------
__global__ __launch_bounds__(256) void edge_kernel(
        const float* __restrict__ hq, const float* __restrict__ hk,
        const int* __restrict__ src, const int* __restrict__ dst,
        float* __restrict__ agg) {
    const int wid  = __builtin_amdgcn_readfirstlane((int)(threadIdx.x >> 5));
    const int e    = blockIdx.x * 8 + wid;   // N_EDGES/8 blocks, exact
    const int lane = threadIdx.x & 31;
    const int c    = lane << 2;              // feature offset 0..124
    const int d = dst[e];                    // wave-uniform -> scalar load
    const int s = src[e];
    const float4 q = *(const float4*)(hq + (size_t)d * DIM + c);
    const float4 k = *(const float4*)(hk + (size_t)s * DIM + c);
    float* a = agg + (size_t)d * DIM + c;
    atomicAdd(a + 0, leaky(q.x + k.x));
    atomicAdd(a + 1, leaky(q.y + k.y));
    atomicAdd(a + 2, leaky(q.z + k.z));
    atomicAdd(a + 3, leaky(q.w + k.w));
}

// ---------------------------------------------------------------------------
// Y[16r:16r+16, :] = epilogue( diag(rowscale) * X @ W + bias )
// 256 threads = 8 waves; block handles one 16-row tile; wave w handles
// columns [16w, 16w+16). A tile staged to LDS via async-to-LDS (ASYNCcnt)
// when available; K-loop: 32x v_wmma_f32_16x16x4_f32 per wave.
// Row scale (1/deg) is applied in the epilogue (commutes with X@W).
// mode: 0 = store, 1 = leaky+store, 2 = accumulate into out.
// ---------------------------------------------------------------------------
__global__ __launch_bounds__(256) void gemm128_kernel(
        const float* __restrict__ X, const float* __restrict__ W,
        const float* __restrict__ bias, const float* __restrict__ rowscale,
        float* __restrict__ out, int mode) {
    __shared__ float As[16 * DIM];   // 8 KB A tile

    const int tid = threadIdx.x;
    const size_t rowBase = (size_t)blockIdx.x * 16;

    // Stage 16x128 f32 A tile into LDS (8 KB): 2x b128 per thread.
    {
        const int i0 = tid * 8;   // 8 floats (32 B) per thread
#if USE_ASYNC_LDS
        __builtin_amdgcn_global_load_async_to_lds_b128(
            (__attribute__((address_space(1))) v4i*)(float*)(X + rowBase * DIM + i0),
            (__attribute__((address_space(3))) v4i*)&As[i0], 0, 0);
        __builtin_amdgcn_global_load_async_to_lds_b128(
            (__attribute__((address_space(1))) v4i*)(float*)(X + rowBase * DIM + i0 + 4),
            (__attribute__((address_space(3))) v4i*)&As[i0 + 4], 0, 0);
        __builtin_amdgcn_s_wait_asynccnt(0);
#else
        const float4* g = (const float4*)(X + rowBase * DIM + i0);
        *(float4*)&As[i0]     = g[0];
        *(float4*)&As[i0 + 4] = g[1];
#endif
    }
    __syncthreads();

    const int lane = tid & 31;
    const int wave = tid >> 5;                  // 0..7 -> column tile
    const int half = lane >> 4;                 // 0 or 1
    const int ncol = (wave << 4) + (lane & 15); // output column
    const int kb = half << 1;                   // K sub-offset {0,2}
    const int arow = lane & 15;                 // A row for this lane

    v8f acc = {};
#pragma unroll
    for (int k0 = 0; k0 < DIM; k0 += 4) {
        // A 16x4: lane holds (M=arow, K=k0+kb, k0+kb+1)
        v2f a = *(const v2f*)&As[arow * DIM + k0 + kb];
        // B 4x16: lane holds (K=k0+kb / +1, N=ncol)
        v2f b;
        b.x = W[(k0 + kb) * DIM + ncol];
        b.y = W[(k0 + kb + 1) * DIM + ncol];
        acc = __builtin_amdgcn_wmma_f32_16x16x4_f32(
                  /*neg_a=*/false, a, /*neg_b=*/false, b,
                  /*c_mod=*/(short)0, acc, /*reuse_a=*/false, /*reuse_b=*/false);
    }

    const float bn = bias[ncol];
    const int mbase = half << 3;                // rows 0..7 or 8..15
#pragma unroll
    for (int j = 0; j < 8; ++j) {
        const size_t row = rowBase + mbase + j;
        float v = acc[j];
        if (rowscale) v *= rowscale[row];       // 1/deg, epilogue-applied
        v += bn;
        if (mode == 1) v = leaky(v);
        else if (mode == 2) v += out[row * DIM + ncol];
        out[row * DIM + ncol] = v;
    }
}

// ---------------------------------------------------------------------------
// Driver
// ---------------------------------------------------------------------------
extern "C" void kernel_launch(void* const* d_in, const int* in_sizes, int n_in,
                              void* d_out, int out_size, void* d_ws, size_t ws_size,
                              hipStream_t stream) {
    const float* feats = (const float*)d_in[0];
    const float* Wq    = (const float*)d_in[1];
    const float* bq    = (const float*)d_in[2];
    const float* Wk    = (const float*)d_in[3];
    const float* bk    = (const float*)d_in[4];
    const float* Wr    = (const float*)d_in[5];
    const float* br    = (const float*)d_in[6];
    const float* Wro   = (const float*)d_in[7];
    const float* bro   = (const float*)d_in[8];
    const int*   src   = (const int*)d_in[9];
    const int*   dst   = (const int*)d_in[10];
    float* out = (float*)d_out;

    const size_t NH = (size_t)N_NODES * DIM;
    float* ws    = (float*)d_ws;
    float* denom = ws;               // N_NODES (holds 1/max(deg,1) after inv)
    float* hq    = denom + N_NODES;  // N x 128
    float* c1    = hq + NH;          // N x 128 (agg1 -> h1)
    float* c2    = c1 + NH;          // N x 128 (agg2 -> h2)
    float* hk    = out;              // reuse output buffer as hk scratch

    const int ROWTILES = N_NODES / 16;      // 3125 (exact)
    const int EDGE_BLOCKS = N_EDGES / 8;    // 100000 (exact, 8 waves/block)

    fill_kernel<<<(N_NODES + 255) / 256, 256, 0, stream>>>(denom, N_NODES, 0.0f);
    fill_kernel<<<((int)NH + 255) / 256, 256, 0, stream>>>(c1, (int)NH, 0.0f);
    fill_kernel<<<((int)NH + 255) / 256, 256, 0, stream>>>(c2, (int)NH, 0.0f);
    deg_kernel<<<(N_EDGES + 255) / 256, 256, 0, stream>>>(dst, denom);
    invdenom_kernel<<<(N_NODES + 255) / 256, 256, 0, stream>>>(denom);

    // ---- layer 0 ----
    gemm128_kernel<<<ROWTILES, 256, 0, stream>>>(feats, Wq, bq, nullptr, hq, 0);
    gemm128_kernel<<<ROWTILES, 256, 0, stream>>>(feats, Wk, bk, nullptr, hk, 0);
    edge_kernel<<<EDGE_BLOCKS, 256, 0, stream>>>(hq, hk, src, dst, c1);
    // h1 = leaky((agg1/deg) @ Wr0 + br0), in place in c1
    gemm128_kernel<<<ROWTILES, 256, 0, stream>>>(c1, Wr, br, denom, c1, 1);

    // ---- layer 1 ----
    gemm128_kernel<<<ROWTILES, 256, 0, stream>>>(c1, Wq + DIM * DIM, bq + DIM, nullptr, hq, 0);
    gemm128_kernel<<<ROWTILES, 256, 0, stream>>>(c1, Wk + DIM * DIM, bk + DIM, nullptr, hk, 0);
    edge_kernel<<<EDGE_BLOCKS, 256, 0, stream>>>(hq, hk, src, dst, c2);
    gemm128_kernel<<<ROWTILES, 256, 0, stream>>>(c2, Wr + DIM * DIM, br + DIM, denom, c2, 1);

    // ---- jumping-knowledge readout: out = sum_i feats_list[i] @ Wro[i] + bro[i]
    gemm128_kernel<<<ROWTILES, 256, 0, stream>>>(feats, Wro, bro, nullptr, out, 0);
    gemm128_kernel<<<ROWTILES, 256, 0, stream>>>(c1, Wro + DIM * DIM, bro + DIM, nullptr, out, 2);
    gemm128_kernel<<<ROWTILES, 256, 0, stream>>>(c2, Wro + 2 * DIM * DIM, bro + 2 * DIM, nullptr, out, 2);
}